// AUGRU_74105365725846
// MI455X (gfx1250) — compile-verified
//
#include <hip/hip_runtime.h>

typedef float v2f __attribute__((ext_vector_type(2)));
typedef float v8f __attribute__((ext_vector_type(8)));

#define TT 50
#define EE 10

// ---------- low-level helpers ----------

__device__ __forceinline__ float bperm(int byteIdx, float v) {
  return __int_as_float(__builtin_amdgcn_ds_bpermute(byteIdx, __float_as_int(v)));
}

// D = A(16x4) * B(4x16) + C, fp32 WMMA
__device__ __forceinline__ v8f wmma4(v2f a, v2f b, v8f c) {
  return __builtin_amdgcn_wmma_f32_16x16x4_f32(false, a, false, b, (short)0, c,
                                               false, false);
}

// full K=12 (E=10 padded) matmul: 3 chained WMMA K-slices
__device__ __forceinline__ v8f mm3(const v2f* A, const v2f* B, v8f c) {
  c = wmma4(A[0], B[0], c);
  c = wmma4(A[1], B[1], c);
  c = wmma4(A[2], B[2], c);
  return c;
}

// Convert a D-layout (16x16, rows=features, cols=batch) tile into the three
// B-operand K-slices.  Rows 10..11 of slice 2 are junk (duplicated rows 8..9)
// but the matching A columns are zero-padded, so they contribute nothing.
__device__ __forceinline__ void d2b(v8f d, int loIdx, bool isLo, v2f* b) {
  float r2 = bperm(loIdx, d[2]);        // row 2 -> hi half of slice0.v0
  float r3 = bperm(loIdx, d[3]);        // row 3 -> hi half of slice0.v1
  float r6 = bperm(loIdx, d[6]);        // row 6 -> hi half of slice1.v0
  float r7 = bperm(loIdx, d[7]);        // row 7 -> hi half of slice1.v1
  float r8 = bperm(loIdx + 64, d[0]);   // row 8 (all lanes)
  float r9 = bperm(loIdx + 64, d[1]);   // row 9 (all lanes)
  b[0] = (v2f){isLo ? d[0] : r2, isLo ? d[1] : r3};
  b[1] = (v2f){isLo ? d[4] : r6, isLo ? d[5] : r7};
  b[2] = (v2f){r8, r9};
}

__device__ __forceinline__ float fast_sigmoid(float x) {
  float e = __builtin_amdgcn_exp2f(-1.4426950408889634f * x);
  return __builtin_amdgcn_rcpf(1.0f + e);
}
__device__ __forceinline__ float fast_tanh(float x) {
  float e = __builtin_amdgcn_exp2f(2.8853900817779268f * x);
  return 1.0f - 2.0f * __builtin_amdgcn_rcpf(e + 1.0f);
}
__device__ __forceinline__ v8f sig8(v8f x) {
  v8f r;
#pragma unroll
  for (int i = 0; i < 8; ++i) r[i] = fast_sigmoid(x[i]);
  return r;
}
__device__ __forceinline__ v8f tanh8(v8f x) {
  v8f r;
#pragma unroll
  for (int i = 0; i < 8; ++i) r[i] = fast_tanh(x[i]);
  return r;
}

// ---------- operand loaders ----------

// A-layout load of W^T, zero-padded to 16x12.  A[m][k] = W[k*E + m].
// lane l holds M = l%16; slice s VGPR v holds K = 4s + v + 2*(l/16).
__device__ __forceinline__ void loadA(const float* __restrict__ W, int m,
                                      int hi, v2f* A) {
#pragma unroll
  for (int s = 0; s < 3; ++s) {
    int k0 = 4 * s + 2 * hi;
    float v0 = 0.0f, v1 = 0.0f;
    if (m < EE) {
      if (k0 < EE) v0 = W[k0 * EE + m];
      if (k0 + 1 < EE) v1 = W[(k0 + 1) * EE + m];
    }
    A[s] = (v2f){v0, v1};
  }
}

// Broadcast a length-E vector across batch columns in D layout (pad rows = 0)
__device__ __forceinline__ v8f loadVecD(const float* __restrict__ b, int hi) {
  v8f r;
#pragma unroll
  for (int v = 0; v < 8; ++v) {
    int f = v + 8 * hi;
    r[v] = (f < EE) ? b[f] : 0.0f;
  }
  return r;
}

// x_t^T as B-operand slices.  xr points at this lane's (batch,t) row; rows
// are 10 floats = 40 bytes so every float2 below is 8-byte aligned.
__device__ __forceinline__ void loadXB(const float* __restrict__ xr, int hi,
                                       v2f* B) {
  float2 p0 = *(const float2*)(xr + 2 * hi);      // K = 2hi, 2hi+1
  float2 p1 = *(const float2*)(xr + 4 + 2 * hi);  // K = 4+2hi, 5+2hi
  float2 p2 = *(const float2*)(xr + 8);           // K = 8,9 (hi half: pad)
  B[0] = (v2f){p0.x, p0.y};
  B[1] = (v2f){p1.x, p1.y};
  B[2] = (v2f){p2.x, p2.y};
}

// a_t^T in D layout (rows=features, cols=batch); pad rows get 0.
__device__ __forceinline__ v8f loadAttD(const float* __restrict__ ar,
                                        bool isLo) {
  v8f r = {0.f, 0.f, 0.f, 0.f, 0.f, 0.f, 0.f, 0.f};
  if (isLo) {
    float2 q0 = *(const float2*)(ar + 0);
    float2 q1 = *(const float2*)(ar + 2);
    float2 q2 = *(const float2*)(ar + 4);
    float2 q3 = *(const float2*)(ar + 6);
    r[0] = q0.x; r[1] = q0.y; r[2] = q1.x; r[3] = q1.y;
    r[4] = q2.x; r[5] = q2.y; r[6] = q3.x; r[7] = q3.y;
  } else {
    float2 q = *(const float2*)(ar + 8);
    r[0] = q.x; r[1] = q.y;
  }
  return r;
}

// ---------- kernel ----------

__global__ __launch_bounds__(256) void augru_wmma_kernel(
    const float* __restrict__ X, const float* __restrict__ Att,
    const float* __restrict__ h0,
    const float* __restrict__ Wi_r, const float* __restrict__ bi_r,
    const float* __restrict__ Wh_r, const float* __restrict__ Ws_r,
    const float* __restrict__ bs_r,
    const float* __restrict__ Wi_z, const float* __restrict__ bi_z,
    const float* __restrict__ Wh_z, const float* __restrict__ Ws_z,
    const float* __restrict__ bs_z,
    const float* __restrict__ Wi_h, const float* __restrict__ bi_h,
    const float* __restrict__ Wh_h, const float* __restrict__ Wt_h,
    const float* __restrict__ bt_h,
    float* __restrict__ out, int nTiles) {
  const int lane = threadIdx.x & 31;
  const int wave = threadIdx.x >> 5;
  const int tile = blockIdx.x * (blockDim.x >> 5) + wave;
  if (tile >= nTiles) return;

  const int col = lane & 15;   // batch column within the 16-wide tile
  const int hi = lane >> 4;    // 0: lanes 0-15, 1: lanes 16-31
  const bool isLo = (hi == 0);
  const int loIdx = col * 4;   // ds_bpermute byte index for "read lane col"
  const int b0 = tile * 16;

  // Preload all 9 weight matrices as WMMA A operands (resident all T steps)
  v2f AWi_r[3], AWh_r[3], AWs_r[3];
  v2f AWi_z[3], AWh_z[3], AWs_z[3];
  v2f AWi_h[3], AWh_h[3], AWt_h[3];
  loadA(Wi_r, col, hi, AWi_r);
  loadA(Wh_r, col, hi, AWh_r);
  loadA(Ws_r, col, hi, AWs_r);
  loadA(Wi_z, col, hi, AWi_z);
  loadA(Wh_z, col, hi, AWh_z);
  loadA(Ws_z, col, hi, AWs_z);
  loadA(Wi_h, col, hi, AWi_h);
  loadA(Wh_h, col, hi, AWh_h);
  loadA(Wt_h, col, hi, AWt_h);

  // Biases broadcast over batch columns, D layout (used as WMMA C operands)
  v8f bir = loadVecD(bi_r, hi), bsr = loadVecD(bs_r, hi);
  v8f biz = loadVecD(bi_z, hi), bsz = loadVecD(bs_z, hi);
  v8f bih = loadVecD(bi_h, hi), bth = loadVecD(bt_h, hi);

  // State h^T (features x 16 batch) in D layout; h0 is (1,E) broadcast
  v8f h = loadVecD(h0, hi);

  const float* xrow = X + (size_t)(b0 + col) * TT * EE;
  const float* arow = Att + (size_t)(b0 + col) * TT * EE;

#pragma unroll 1
  for (int t = 0; t < TT; ++t, xrow += EE, arow += EE) {
    // stream inputs a few steps ahead (whole dataset > L2)
    __builtin_prefetch(xrow + 8 * EE, 0, 1);
    __builtin_prefetch(arow + 8 * EE, 0, 1);

    v2f xB[3];
    loadXB(xrow, hi, xB);
    v8f aD = loadAttD(arow, isLo);

    // u_* = Wi_*^T @ x_t^T + bi_*  (bias as C operand)
    v8f u_r = mm3(AWi_r, xB, bir);
    v8f u_z = mm3(AWi_z, xB, biz);
    v8f u_h = mm3(AWi_h, xB, bih);

    // tmp_r/z = u + Wh^T @ h^T
    v2f hB[3];
    d2b(h, loIdx, isLo, hB);
    v8f tr = mm3(AWh_r, hB, u_r);
    v8f tz = mm3(AWh_z, hB, u_z);

    // r = sigmoid(Ws_r^T @ tmp_r + bs_r)
    v2f trB[3];
    d2b(tr, loIdx, isLo, trB);
    v8f r = sig8(mm3(AWs_r, trB, bsr));

    // z = sigmoid(Ws_z^T @ tmp_z + bs_z)
    v2f tzB[3];
    d2b(tz, loIdx, isLo, tzB);
    v8f z = sig8(mm3(AWs_z, tzB, bsz));

    // hc = tanh(Wt_h^T @ (u_h + Wh_h^T @ (h*z)) + bt_h)
    v8f hz = h * z;
    v2f hzB[3];
    d2b(hz, loIdx, isLo, hzB);
    v8f th = mm3(AWh_h, hzB, u_h);
    v2f thB[3];
    d2b(th, loIdx, isLo, thB);
    v8f hc = tanh8(mm3(AWt_h, thB, bth));

    // h = (1 - a*r) * h + (a*r) * hc
    v8f Ra = aD * r;
    h = h + Ra * (hc - h);
  }

  // Store h^T back as row-major (B, E).  Lane's column row base is 8B aligned.
  float* orow = out + (size_t)(b0 + col) * EE;
  if (isLo) {
    *(float2*)(orow + 0) = make_float2(h[0], h[1]);
    *(float2*)(orow + 2) = make_float2(h[2], h[3]);
    *(float2*)(orow + 4) = make_float2(h[4], h[5]);
    *(float2*)(orow + 6) = make_float2(h[6], h[7]);
  } else {
    *(float2*)(orow + 8) = make_float2(h[0], h[1]);  // features 8,9
  }
}

// ---------- launch ----------

extern "C" void kernel_launch(void* const* d_in, const int* in_sizes, int n_in,
                              void* d_out, int out_size, void* d_ws,
                              size_t ws_size, hipStream_t stream) {
  const float* X    = (const float*)d_in[0];
  const float* Att  = (const float*)d_in[1];
  const float* h0   = (const float*)d_in[2];
  const float* Wi_r = (const float*)d_in[3];
  const float* bi_r = (const float*)d_in[4];
  const float* Wh_r = (const float*)d_in[5];
  const float* Ws_r = (const float*)d_in[6];
  const float* bs_r = (const float*)d_in[7];
  const float* Wi_z = (const float*)d_in[8];
  const float* bi_z = (const float*)d_in[9];
  const float* Wh_z = (const float*)d_in[10];
  const float* Ws_z = (const float*)d_in[11];
  const float* bs_z = (const float*)d_in[12];
  const float* Wi_h = (const float*)d_in[13];
  const float* bi_h = (const float*)d_in[14];
  const float* Wh_h = (const float*)d_in[15];
  const float* Wt_h = (const float*)d_in[16];
  const float* bt_h = (const float*)d_in[17];

  const int B = in_sizes[0] / (TT * EE);   // 65536
  const int nTiles = B / 16;               // 4096 wave-tiles
  const int wavesPerBlock = 8;             // 256 threads = 8 wave32
  dim3 block(32 * wavesPerBlock);
  dim3 grid((nTiles + wavesPerBlock - 1) / wavesPerBlock);

  augru_wmma_kernel<<<grid, block, 0, stream>>>(
      X, Att, h0, Wi_r, bi_r, Wh_r, Ws_r, bs_r, Wi_z, bi_z, Wh_z, Ws_z, bs_z,
      Wi_h, bi_h, Wh_h, Wt_h, bt_h, (float*)d_out, nTiles);
}